// NeRFAccSampler_55791625175295
// MI455X (gfx1250) — compile-verified
//
#include <hip/hip_runtime.h>

// NeRF-acc style occupancy-grid ray marcher for MI455X (gfx1250, wave32).
//
// Roofline: 4 outputs * 16384 rays * 1024 samples * 4B = 268 MB stores
// -> ~11.5us floor at 23.3 TB/s. Compute (~0.4 GFLOP) is negligible; the
// only serial dependence (per-ray transmittance cumsum) is turned into
// chained V_WMMA_F32_16X16X4_F32 ops (scan == matmul with a triangular
// ones matrix), keeping full f32 precision in the accumulator.

#define RES     128
#define SSTEPS  1024
#define NRAYS   16384
#define WAVES_PER_BLOCK 8
// STEP = 2*sqrt(3)/1024
#define STEPC   (3.4641016151377544f / 1024.0f)
// T > 1e-4  <=>  tau_excl < -ln(1e-4)
#define TAU_MAX 9.210340371976184f

typedef __attribute__((ext_vector_type(2))) float v2f;
typedef __attribute__((ext_vector_type(8))) float v8f;

__global__ __launch_bounds__(256) void nerfacc_march_kernel(
    const float* __restrict__ rays_o,
    const float* __restrict__ rays_d,
    const float* __restrict__ jitter,
    const float* __restrict__ Wd,
    const float* __restrict__ bd,
    const int*   __restrict__ binaries,
    float*       __restrict__ out)
{
    const int lane = threadIdx.x & 31;
    const int wib  = threadIdx.x >> 5;                 // wave in block
    const int waveGlobal = blockIdx.x * WAVES_PER_BLOCK + wib;
    const int rayBase = waveGlobal * 16;               // 16 rays per wave
    const int rl   = lane & 15;                        // A-layout row / D-layout col
    const int half = lane >> 4;                        // 0 or 1

    // Per-wave 16x16 transpose tile (A-layout producer -> D-layout consumer)
    __shared__ float lds_tile[WAVES_PER_BLOCK][16][16];
    float (*tile)[16] = lds_tile[wib];

    // ---- per-ray setup (this lane's A-layout ray; both halves duplicate) ----
    const int ray = rayBase + rl;
    const float ox = rays_o[ray * 3 + 0];
    const float oy = rays_o[ray * 3 + 1];
    const float oz = rays_o[ray * 3 + 2];
    const float dx = rays_d[ray * 3 + 0];
    const float dy = rays_d[ray * 3 + 1];
    const float dz = rays_d[ray * 3 + 2];
    const float jit = jitter[ray];

    const float w0 = Wd[0], w1 = Wd[1], w2 = Wd[2], bb = bd[0];

    // AABB [-1,1]^3 slab test
    const float ix_ = 1.0f / dx, iy_ = 1.0f / dy, iz_ = 1.0f / dz;
    const float tx0 = (-1.0f - ox) * ix_, tx1 = (1.0f - ox) * ix_;
    const float ty0 = (-1.0f - oy) * iy_, ty1 = (1.0f - oy) * iy_;
    const float tz0 = (-1.0f - oz) * iz_, tz1 = (1.0f - oz) * iz_;
    const float tnr = fmaxf(fmaxf(fminf(tx0, tx1), fminf(ty0, ty1)), fminf(tz0, tz1));
    const float tfr = fminf(fminf(fmaxf(tx0, tx1), fmaxf(ty0, ty1)), fmaxf(tz0, tz1));
    const float t_near = fmaxf(tnr, 0.0f);
    const float t_far  = tfr;
    const bool  hit    = t_near < t_far;

    // ---- hoisted row broadcasts for D-layout consumption (rows v + 8*half) ----
    float tnRow[8], jitRow[8];
#pragma unroll
    for (int v = 0; v < 8; ++v) {
        const int r = v + 8 * half;
        tnRow[v]  = __shfl(t_near, r, 32);
        jitRow[v] = __shfl(jit,    r, 32);
    }

    // ---- triangular-ones B constants: B[k][col] = (k_global <= col) ----
    // B 4x16 f32 layout assumed mirror of A: VGPR e holds row K = e + 2*half,
    // col = lane%16.
    v2f Bc[4];
#pragma unroll
    for (int c = 0; c < 4; ++c) {
        Bc[c].x = ((4 * c + 0 + 2 * half) <= rl) ? 1.0f : 0.0f;
        Bc[c].y = ((4 * c + 1 + 2 * half) <= rl) ? 1.0f : 0.0f;
    }

    float carry[8];
#pragma unroll
    for (int v = 0; v < 8; ++v) carry[v] = 0.0f;

    constexpr long TOT = (long)NRAYS * (long)SSTEPS;

    for (int chunk = 0; chunk < SSTEPS / 16; ++chunk) {
        const int base = chunk * 16;

        // ---- produce sigma*STEP in A layout; stash (valid, sigma*STEP) to LDS ----
        v2f A[4];
#pragma unroll
        for (int c = 0; c < 4; ++c) {
#pragma unroll
            for (int e = 0; e < 2; ++e) {
                const int j = base + 4 * c + e + 2 * half;   // sample index
                const float ts = t_near + ((float)j + jit) * STEPC;
                const float tm = ts + 0.5f * STEPC;
                const float px = fmaf(dx, tm, ox);
                const float py = fmaf(dy, tm, oy);
                const float pz = fmaf(dz, tm, oz);
                // grid cell: clip(floor((p+1)*64), 0, 127)
                const int gx = (int)fminf(fmaxf(floorf((px + 1.0f) * 64.0f), 0.0f), 127.0f);
                const int gy = (int)fminf(fmaxf(floorf((py + 1.0f) * 64.0f), 0.0f), 127.0f);
                const int gz = (int)fminf(fmaxf(floorf((pz + 1.0f) * 64.0f), 0.0f), 127.0f);
                const int flat = (gx * RES + gy) * RES + gz;
                const bool occ = binaries[flat] != 0;
                const bool inb = hit && (ts + STEPC < t_far);
                const bool valid = occ && inb;
                const float dens = fmaxf(fmaf(px, w0, fmaf(py, w1, fmaf(pz, w2, bb))), 0.0f);
                const float sST = valid ? dens * STEPC : 0.0f;
                if (e == 0) A[c].x = sST; else A[c].y = sST;
                tile[rl][4 * c + e + 2 * half] = valid ? sST : -1.0f;
            }
        }

#if defined(__has_builtin)
#if __has_builtin(__builtin_amdgcn_sched_barrier)
        __builtin_amdgcn_sched_barrier(0);
#endif
#endif

        // ---- inclusive 16-wide prefix scan as 4 chained f32 WMMAs ----
        v8f acc = (v8f){0.f, 0.f, 0.f, 0.f, 0.f, 0.f, 0.f, 0.f};
#pragma unroll
        for (int c = 0; c < 4; ++c) {
            acc = __builtin_amdgcn_wmma_f32_16x16x4_f32(
                /*neg_a=*/false, A[c], /*neg_b=*/false, Bc[c],
                /*c_mod=*/(short)0, acc, /*reuse_a=*/false, /*reuse_b=*/false);
        }

        // ---- consume in D layout: row = v + 8*half, col = rl ----
#pragma unroll
        for (int v = 0; v < 8; ++v) {
            const int r = v + 8 * half;
            const float incl = acc[v];
            const float enc  = tile[r][rl];
            const bool  valid = enc >= 0.0f;
            const float sST   = fmaxf(enc, 0.0f);
            const float tau_excl = carry[v] + incl - sST;
            const bool  keep = valid && (tau_excl < TAU_MAX);

            const float ts = tnRow[v] + ((float)(base + rl) + jitRow[v]) * STEPC;
            const int   ray_r = rayBase + r;
            const long  o0 = (long)ray_r * SSTEPS + (base + rl);

            out[o0]            = (float)ray_r;                 // ray_indices
            out[o0 + TOT]      = keep ? ts : 0.0f;             // t_starts * keep
            out[o0 + 2 * TOT]  = keep ? ts + STEPC : 0.0f;     // t_ends * keep
            out[o0 + 3 * TOT]  = keep ? 1.0f : 0.0f;           // keep

            // advance running tau: broadcast inclusive sum at col 15 of this half
            carry[v] += __shfl(incl, 15 + 16 * half, 32);
        }
    }
}

extern "C" void kernel_launch(void* const* d_in, const int* in_sizes, int n_in,
                              void* d_out, int out_size, void* d_ws, size_t ws_size,
                              hipStream_t stream) {
    (void)in_sizes; (void)n_in; (void)out_size; (void)d_ws; (void)ws_size;
    const float* rays_o   = (const float*)d_in[0];
    const float* rays_d   = (const float*)d_in[1];
    const float* jitter   = (const float*)d_in[2];
    const float* Wd       = (const float*)d_in[3];
    const float* bd       = (const float*)d_in[4];
    const int*   binaries = (const int*)d_in[5];
    float* out = (float*)d_out;

    // 16384 rays / 16 rays-per-wave = 1024 waves; 8 waves (256 thr) per block.
    const int blocks = (NRAYS / 16) / WAVES_PER_BLOCK;   // 128
    nerfacc_march_kernel<<<blocks, 256, 0, stream>>>(
        rays_o, rays_d, jitter, Wd, bd, binaries, out);
}